// RasterizeRGB_15625091023490
// MI455X (gfx1250) — compile-verified
//
#include <hip/hip_runtime.h>
#include <math.h>

typedef __attribute__((ext_vector_type(2))) float v2f;
typedef __attribute__((ext_vector_type(4))) float v4f;
typedef __attribute__((ext_vector_type(8))) float v8f;

#define IMG      256
#define NPIX     (IMG * IMG)        // 65536
#define F_CNT    256
#define TSZ      4
#define NEAR_P   0.1f
#define FAR_P    100.0f
#define EPS_T    0.001f

// Workspace layout (floats):
//   P0 @ 0    : B0 WMMA operand pack, [chunk=16][lane=32][2]   = 1024
//   P1 @ 1024 : B1 WMMA operand pack, [chunk=16][lane=32][2]   = 1024
//   P2 @ 2048 : per-face {1/z0,1/z1,1/z2,0}                    = 1024
//   P3 @ 3072 : per-face {a0,b0,c0,a1,b1,c1,0,0}               = 2048
#define WS_P0 0
#define WS_P1 1024
#define WS_P2 2048
#define WS_P3 3072

// ---------------------------------------------------------------------------
// Kernel 1: fold faces into affine barycentric coefficients, pre-packed in
// the v_wmma_f32_16x16x4_f32 B-operand lane layout
// (VGPR0: K=0 lanes 0-15 / K=2 lanes 16-31; VGPR1: K=1 / K=3).
// ---------------------------------------------------------------------------
__global__ void RasterizeRGB_facepre_kernel(const float* __restrict__ faces,
                                            float* __restrict__ ws) {
  int f = blockIdx.x * blockDim.x + threadIdx.x;
  if (f >= F_CNT) return;
  const float* v = faces + f * 9;
  float x0 = v[0], y0 = v[1], z0 = v[2];
  float x1 = v[3], y1 = v[4], z1 = v[5];
  float x2 = v[6], y2 = v[7], z2 = v[8];

  float denom = (y1 - y2) * (x0 - x2) + (x2 - x1) * (y0 - y2);
  if (fabsf(denom) < 1e-12f) denom = 1e-12f;
  float inv_d = 1.0f / denom;

  float a0 = (y1 - y2) * inv_d;
  float b0 = (x2 - x1) * inv_d;
  float a1 = (y2 - y0) * inv_d;
  float b1 = (x0 - x2) * inv_d;
  float c0 = -a0 * x2 - b0 * y2;   // w0 = a0*px + b0*py + c0
  float c1 = -a1 * x2 - b1 * y2;   // w1 = a1*px + b1*py + c1

  const int ch  = f >> 4;
  const int col = f & 15;

  // B0/B1 operand pack: lanes 0-15 hold (a,b) rows K=0,1; lanes 16-31 hold
  // (c,0) rows K=2,3 for the same face column.
  float* p0 = ws + WS_P0;
  float* p1 = ws + WS_P1;
  p0[(ch * 32 + col) * 2 + 0]      = a0;
  p0[(ch * 32 + col) * 2 + 1]      = b0;
  p0[(ch * 32 + 16 + col) * 2 + 0] = c0;
  p0[(ch * 32 + 16 + col) * 2 + 1] = 0.0f;
  p1[(ch * 32 + col) * 2 + 0]      = a1;
  p1[(ch * 32 + col) * 2 + 1]      = b1;
  p1[(ch * 32 + 16 + col) * 2 + 0] = c1;
  p1[(ch * 32 + 16 + col) * 2 + 1] = 0.0f;

  float* p2 = ws + WS_P2 + f * 4;
  p2[0] = 1.0f / z0;
  p2[1] = 1.0f / z1;
  p2[2] = 1.0f / z2;
  p2[3] = 0.0f;

  float* p3 = ws + WS_P3 + f * 8;
  p3[0] = a0; p3[1] = b0; p3[2] = c0;
  p3[3] = a1; p3[4] = b1; p3[5] = c1;
  p3[6] = 0.0f; p3[7] = 0.0f;
}

// ---------------------------------------------------------------------------
// Kernel 2: WMMA rasterizer. Block = 256 threads = 8 wave32; each wave owns a
// 16-pixel strip (same image row). Per 16-face chunk: two
// v_wmma_f32_16x16x4_f32 give w0,w1 for the 16x16 (pixel x face) tile; VALU
// does clip/renorm/perspective-z and a packed-u64-key z-buffer argmin.
// ---------------------------------------------------------------------------
__global__ __launch_bounds__(256) void RasterizeRGB_raster_kernel(
    const float* __restrict__ ws,
    const float* __restrict__ textures,
    float* __restrict__ out) {
  __shared__ unsigned long long s_k[128];

  const int tid   = threadIdx.x;
  const int lane  = tid & 31;
  const int wave  = tid >> 5;
  const int blockPix = blockIdx.x * 128;          // 128 pixels per block
  const int strip = blockPix + wave * 16;         // 16 consecutive pixels
  const int col0  = strip & (IMG - 1);
  const int row   = strip >> 8;
  const int mcol  = lane & 15;
  const bool hi   = lane >= 16;

  const float* p0 = ws + WS_P0;
  const float* p1 = ws + WS_P1;
  const float* p2 = ws + WS_P2;
  const float* p3 = ws + WS_P3;

  // A operand [16x4] f32: rows M=pixel, K = (px, py, 1, 0).
  const float px = (2.0f * (float)(col0 + mcol) + 1.0f - (float)IMG) / (float)IMG;
  const float py = -((2.0f * (float)row + 1.0f - (float)IMG) / (float)IMG);
  v2f A;
  A.x = hi ? 1.0f : px;
  A.y = hi ? 0.0f : py;

  const unsigned long long KINF = 0x7f800000ull << 32;  // +inf, face 0
  unsigned long long kmin[8];
#pragma unroll
  for (int v = 0; v < 8; ++v) kmin[v] = KINF;

  for (int ch = 0; ch < F_CNT / 16; ++ch) {
    const unsigned n = (unsigned)(ch * 16 + mcol);  // this lane's face column

    v2f B0 = *(const v2f*)(p0 + (ch * 32 + lane) * 2);  // b64, coalesced
    v2f B1 = *(const v2f*)(p1 + (ch * 32 + lane) * 2);
    v4f izv = *(const v4f*)(p2 + n * 4);                // b128

    v8f czero = {};
    v8f W0 = __builtin_amdgcn_wmma_f32_16x16x4_f32(
        false, A, false, B0, (short)0, czero, false, false);
    v8f W1 = __builtin_amdgcn_wmma_f32_16x16x4_f32(
        false, A, false, B1, (short)0, czero, false, false);

    const float i0 = izv.x, i1 = izv.y, i2 = izv.z;

#pragma unroll
    for (int v = 0; v < 8; ++v) {
      float w0 = W0[v];
      float w1 = W1[v];
      float w2 = 1.0f - w0 - w1;
      float mn = fminf(fminf(w0, w1), w2);

      float cc0 = fminf(fmaxf(w0, 0.0f), 1.0f);
      float cc1 = fminf(fmaxf(w1, 0.0f), 1.0f);
      float cc2 = fminf(fmaxf(w2, 0.0f), 1.0f);
      float rs  = __builtin_amdgcn_rcpf(fmaxf(cc0 + cc1 + cc2, 1e-12f));
      float dot = cc0 * i0 + cc1 * i1 + cc2 * i2;
      float invz = dot * rs;
      float zp = __builtin_copysignf(
          __builtin_amdgcn_rcpf(fmaxf(fabsf(invz), 1e-12f)), invz + 1e-30f);

      bool valid = (mn >= 0.0f) && (zp > NEAR_P) && (zp < FAR_P);
      float zb = valid ? zp : __builtin_inff();
      // zb > 0 always when valid -> IEEE bits are order-isomorphic to value.
      unsigned long long key =
          ((unsigned long long)__float_as_uint(zb) << 32) | n;
      kmin[v] = (key < kmin[v]) ? key : kmin[v];  // ties -> smaller face idx
    }
  }

  // Cross-lane argmin within each 16-lane half (masks stay inside the half).
#pragma unroll
  for (int v = 0; v < 8; ++v) {
    for (int m = 8; m > 0; m >>= 1) {
      unsigned long long ko = __shfl_xor(kmin[v], m, 32);
      kmin[v] = (ko < kmin[v]) ? ko : kmin[v];
    }
  }

  // Stage per-pixel winners to LDS: pixel M = v + 8*(lane>=16).
  unsigned long long ksel = kmin[0];
#pragma unroll
  for (int v = 1; v < 8; ++v) {
    bool pick = ((lane & 7) == v);
    ksel = pick ? kmin[v] : ksel;
  }
  if ((lane & 15) < 8) {
    int pix = wave * 16 + (hi ? 8 : 0) + (lane & 7);
    s_k[pix] = ksel;
  }
  __syncthreads();

  // Shading: one thread per pixel (128 pixels per block).
  if (tid < 128) {
    const int p = blockPix + tid;
    const unsigned long long k = s_k[tid];
    const unsigned zbits = (unsigned)(k >> 32);
    float r = 0.0f, g = 0.0f, b = 0.0f;

    if (zbits < 0x7f800000u) {  // finite -> hit
      const float zp = __uint_as_float(zbits);
      const int f = (int)(k & 0xffffffffu);

      const int pcol = p & (IMG - 1);
      const int prow = p >> 8;
      const float ppx = (2.0f * (float)pcol + 1.0f - (float)IMG) / (float)IMG;
      const float ppy = -((2.0f * (float)prow + 1.0f - (float)IMG) / (float)IMG);

      v4f cf0 = *(const v4f*)(p3 + f * 8);      // a0,b0,c0,a1
      v4f cf1 = *(const v4f*)(p3 + f * 8 + 4);  // b1,c1,-,-
      v4f izv = *(const v4f*)(p2 + f * 4);      // 1/z0,1/z1,1/z2

      float w0 = cf0.x * ppx + cf0.y * ppy + cf0.z;
      float w1 = cf0.w * ppx + cf1.x * ppy + cf1.y;
      float w2 = 1.0f - w0 - w1;
      float cc0 = fminf(fmaxf(w0, 0.0f), 1.0f);
      float cc1 = fminf(fmaxf(w1, 0.0f), 1.0f);
      float cc2 = fminf(fmaxf(w2, 0.0f), 1.0f);
      float rs = __builtin_amdgcn_rcpf(fmaxf(cc0 + cc1 + cc2, 1e-12f));
      cc0 *= rs; cc1 *= rs; cc2 *= rs;

      const float kTmax = (float)(TSZ - 1) - EPS_T;
      float tf0 = fminf(fmaxf(cc0 * (float)(TSZ - 1) * zp * izv.x, 0.0f), kTmax);
      float tf1 = fminf(fmaxf(cc1 * (float)(TSZ - 1) * zp * izv.y, 0.0f), kTmax);
      float tf2 = fminf(fmaxf(cc2 * (float)(TSZ - 1) * zp * izv.z, 0.0f), kTmax);
      int ti0 = (int)floorf(tf0);
      int ti1 = (int)floorf(tf1);
      int ti2 = (int)floorf(tf2);
      float fr0 = tf0 - (float)ti0;
      float fr1 = tf1 - (float)ti1;
      float fr2 = tf2 - (float)ti2;

      const float* tex = textures + (size_t)f * (TSZ * TSZ * TSZ * 3);
#pragma unroll
      for (int corner = 0; corner < 8; ++corner) {
        int bb0 = corner & 1;
        int bb1 = (corner >> 1) & 1;
        int bb2 = (corner >> 2) & 1;
        float w = (bb0 ? fr0 : 1.0f - fr0) *
                  (bb1 ? fr1 : 1.0f - fr1) *
                  (bb2 ? fr2 : 1.0f - fr2);
        int i0 = min(ti0 + bb0, TSZ - 1);
        int i1 = min(ti1 + bb1, TSZ - 1);
        int i2 = min(ti2 + bb2, TSZ - 1);
        const float* t = tex + ((i0 * TSZ + i1) * TSZ + i2) * 3;
        r += w * t[0];
        g += w * t[1];
        b += w * t[2];
      }
    }
    out[0 * NPIX + p] = r;
    out[1 * NPIX + p] = g;
    out[2 * NPIX + p] = b;
  }
}

extern "C" void kernel_launch(void* const* d_in, const int* in_sizes, int n_in,
                              void* d_out, int out_size, void* d_ws, size_t ws_size,
                              hipStream_t stream) {
  (void)in_sizes; (void)n_in; (void)out_size; (void)ws_size;
  const float* faces    = (const float*)d_in[0];  // [1,256,3,3]
  const float* textures = (const float*)d_in[1];  // [1,256,4,4,4,3]
  float* out = (float*)d_out;                     // [1,3,256,256]
  float* ws  = (float*)d_ws;                      // 5120 floats

  RasterizeRGB_facepre_kernel<<<1, 256, 0, stream>>>(faces, ws);
  RasterizeRGB_raster_kernel<<<NPIX / 128, 256, 0, stream>>>(ws, textures, out);
}